// BaseGraphPooling_69810398429540
// MI455X (gfx1250) — compile-verified
//
#include <hip/hip_runtime.h>

typedef __attribute__((ext_vector_type(2))) float v2f;
typedef __attribute__((ext_vector_type(8))) float v8f;

#define D_FEAT 128

// ---------------- zero the sum buffer (d_out) ----------------
__global__ void zero_kernel(float* __restrict__ out, int total) {
  int i = blockIdx.x * blockDim.x + threadIdx.x;
  int stride = gridDim.x * blockDim.x;
  for (; i < total; i += stride) out[i] = 0.0f;
}

// ---------------- exact counts via binary search on sorted ids ----------------
__global__ void counts_kernel(const int* __restrict__ batch, float* __restrict__ counts,
                              int N, int G) {
  int g = blockIdx.x * blockDim.x + threadIdx.x;
  if (g >= G) return;
  int lo = 0, hi = N;                       // lower_bound(g)
  while (lo < hi) { int mid = (lo + hi) >> 1; if (batch[mid] < g) lo = mid + 1; else hi = mid; }
  int lo2 = lo, hi2 = N;                    // lower_bound(g+1)
  while (lo2 < hi2) { int mid = (lo2 + hi2) >> 1; if (batch[mid] < g + 1) lo2 = mid + 1; else hi2 = mid; }
  counts[g] = (float)(lo2 - lo);
}

// ---------------- WMMA segmented sum ----------------
// Block = 8 waves; wave w owns features [16w,16w+16) of its block's node chunk.
// Fast path: 16 nodes / iteration -> 4x v_wmma_f32_16x16x4_f32, one uniform
// window check, and pointer-increment addressing so all loads use immediate
// offsets. Slow path (id jump >= 16, chunk tail) stays fully general but cold.
__global__ void segsum_wmma_kernel(const float* __restrict__ node_h,
                                   const int* __restrict__ batch,
                                   float* __restrict__ sums,
                                   int N) {
  const int lane  = threadIdx.x & 31;
  const int wave  = threadIdx.x >> 5;   // 0..7
  const int fbase = wave * 16;          // this wave's feature slice
  const int half  = lane >> 4;          // 0 | 1
  const int col   = lane & 15;          // N-index for B/C, M-index for A
  const int kA    = 2 * half;           // node slot held in vgpr0 (K=0 / K=2)
  const int kB    = 2 * half + 1;       // node slot held in vgpr1 (K=1 / K=3)

  const int chunk = (N + gridDim.x - 1) / gridDim.x;
  const int begin = blockIdx.x * chunk;
  const int end   = min(begin + chunk, N);
  if (begin >= end) return;

  v8f c = {};
  int wbase  = __builtin_amdgcn_readfirstlane(batch[begin]);  // window base segment
  int maxloc = -1;                                            // max local seg touched

  auto flush = [&]() {
#pragma unroll
    for (int v = 0; v < 8; ++v) {
      int m = v + 8 * half;             // C/D layout: lanes 0-15 hold M=v, 16-31 M=v+8
      if (m <= maxloc)
        atomicAdd(sums + (size_t)(wbase + m) * D_FEAT + fbase + col, c[v]);
      c[v] = 0.0f;
    }
    maxloc = -1;
  };

  // Fully general 4-node tile (cold path).
  auto process4 = [&](int i4) {
    int  s[4];
    bool val[4];
#pragma unroll
    for (int k = 0; k < 4; ++k) {
      int idx = i4 + k;
      val[k] = idx < end;
      s[k]   = __builtin_amdgcn_readfirstlane(batch[min(idx, N - 1)]);
    }
    float b0 = val[kA] ? node_h[(size_t)(i4 + kA) * D_FEAT + fbase + col] : 0.0f;
    float b1 = val[kB] ? node_h[(size_t)(i4 + kB) * D_FEAT + fbase + col] : 0.0f;
    v2f b = {b0, b1};
    int p = 0;
    while (p < 4 && val[p]) {           // wave-uniform
      if (s[p] - wbase >= 16) { flush(); wbase = s[p]; }
      bool incA = (kA >= p) && val[kA] && (s[kA] - wbase < 16);
      bool incB = (kB >= p) && val[kB] && (s[kB] - wbase < 16);
      float a0 = (incA && (s[kA] - wbase == col)) ? 1.0f : 0.0f;
      float a1 = (incB && (s[kB] - wbase == col)) ? 1.0f : 0.0f;
      v2f a = {a0, a1};
      c = __builtin_amdgcn_wmma_f32_16x16x4_f32(false, a, false, b, (short)0, c,
                                                false, false);
      int q = p;
      while (q < 4 && val[q] && (s[q] - wbase < 16)) ++q;
      maxloc = max(maxloc, s[q - 1] - wbase);
      p = q;
    }
  };

  // Per-lane carried pointers: feature loads become {0,512,2048,...} immediate
  // offsets off hp; the 4 int2 id loads become {0,16,32,48} offsets off bp.
  const float* hp = node_h + (size_t)(begin + kA) * D_FEAT + fbase + col;
  const int*   bp = batch + (begin + kA);

  int i = begin;
  while (i + 16 <= end) {
    // Segment ids for this lane's 8 nodes, loaded exactly in WMMA A layout:
    // group j: .x -> node i+4j+kA (A vgpr0), .y -> node i+4j+kB (A vgpr1)
    int2 sp[4];
#pragma unroll
    for (int j = 0; j < 4; ++j)
      sp[j] = *(const int2*)(bp + 4 * j);

    int sfirst = __builtin_amdgcn_readlane(sp[0].x, 0);   // id of node i
    int slast  = __builtin_amdgcn_readlane(sp[3].y, 31);  // id of node i+15

    if (slast - wbase >= 16 && slast - sfirst < 16) {     // uniform rebase
      flush();
      wbase = sfirst;
    }

    if (slast - wbase < 16) {
      // ---------- fast path: 16 nodes, 4 WMMAs, no inner control flow ----------
      if (i + 80 <= end)                                  // stream ~32KB ahead
        __builtin_prefetch(hp + (size_t)64 * D_FEAT, 0, 1);
#pragma unroll
      for (int j = 0; j < 4; ++j) {
        float b0 = hp[(size_t)(4 * j) * D_FEAT];          // node i+4j+kA
        float b1 = hp[(size_t)(4 * j + 1) * D_FEAT];      // node i+4j+kB
        float a0 = (sp[j].x - wbase == col) ? 1.0f : 0.0f;
        float a1 = (sp[j].y - wbase == col) ? 1.0f : 0.0f;
        v2f a = {a0, a1};
        v2f b = {b0, b1};
        c = __builtin_amdgcn_wmma_f32_16x16x4_f32(false, a, false, b, (short)0, c,
                                                  false, false);
      }
      maxloc = max(maxloc, slast - wbase);  // ids sorted -> uniform running max
    } else {
      // ---------- pathological id jump inside 16 nodes (rare) ----------
      process4(i); process4(i + 4); process4(i + 8); process4(i + 12);
    }
    i += 16;
    hp += (size_t)16 * D_FEAT;
    bp += 16;
  }
  for (; i < end; i += 4) process4(i);      // chunk tail
  flush();
}

// ---------------- divide by clamped counts ----------------
__global__ void finalize_kernel(float* __restrict__ out,
                                const float* __restrict__ counts, int total) {
  int i = blockIdx.x * blockDim.x + threadIdx.x;
  int stride = gridDim.x * blockDim.x;
  for (; i < total; i += stride) {
    int g = i / D_FEAT;
    out[i] = out[i] / fmaxf(counts[g], 1.0f);
  }
}

extern "C" void kernel_launch(void* const* d_in, const int* in_sizes, int n_in,
                              void* d_out, int out_size, void* d_ws, size_t ws_size,
                              hipStream_t stream) {
  const float* node_h = (const float*)d_in[0];
  const int*   batch  = (const int*)d_in[1];
  int N = in_sizes[1];                 // number of nodes
  int G = out_size / D_FEAT;           // number of graphs
  float* out    = (float*)d_out;       // G x 128 sums -> means
  float* counts = (float*)d_ws;        // G floats of scratch

  zero_kernel<<<(out_size + 255) / 256, 256, 0, stream>>>(out, out_size);
  counts_kernel<<<(G + 255) / 256, 256, 0, stream>>>(batch, counts, N, G);
  segsum_wmma_kernel<<<1024, 256, 0, stream>>>(node_h, batch, out, N);
  finalize_kernel<<<(out_size + 255) / 256, 256, 0, stream>>>(out, counts, out_size);
}